// EfficientMemorySoftmax_45981919870994
// MI455X (gfx1250) — compile-verified
//
#include <hip/hip_runtime.h>
#include <stdint.h>

// Softmax over rows of length 1024, fp32 in / fp32 out.
// Shape: (4,16,1024,1024) -> 65536 rows x 1024.
// Pure bandwidth-bound: 512 MiB of HBM traffic, ~23 us floor at 23.3 TB/s.
// Strategy: one wave32 per row; row staged global->LDS via CDNA5 async
// loads; all reductions are intra-wave shuffles (no barriers); NT stores.

#define ROW_LEN        1024
#define WAVES_PER_BLK  8
#define THREADS        (WAVES_PER_BLK * 32)
#define CHUNKS         (ROW_LEN / (32 * 4))   // 8 float4 chunks per lane

typedef __attribute__((ext_vector_type(4))) float v4f;
typedef int v4i __attribute__((vector_size(4 * sizeof(int))));
typedef __attribute__((address_space(1))) v4i* gv4i_ptr;
typedef __attribute__((address_space(3))) v4i* lv4i_ptr;

__device__ __forceinline__ float wave_max(float v) {
#pragma unroll
  for (int m = 16; m >= 1; m >>= 1)
    v = fmaxf(v, __shfl_xor(v, m, 32));
  return v;
}

__device__ __forceinline__ float wave_sum(float v) {
#pragma unroll
  for (int m = 16; m >= 1; m >>= 1)
    v += __shfl_xor(v, m, 32);
  return v;
}

__global__ __launch_bounds__(THREADS, 1) void softmax1024_kernel(
    const float* __restrict__ x, float* __restrict__ out, int nrows) {
#if __has_builtin(__builtin_amdgcn_global_load_async_to_lds_b128)
  __shared__ float smem[WAVES_PER_BLK * ROW_LEN];
#endif
  const int wave = threadIdx.x >> 5;
  const int lane = threadIdx.x & 31;
  const int row  = blockIdx.x * WAVES_PER_BLK + wave;
  if (row >= nrows) return;

  const float* __restrict__ rowp = x   + (size_t)row * ROW_LEN;
  float*       __restrict__ orow = out + (size_t)row * ROW_LEN;

  v4f vals[CHUNKS];

#if __has_builtin(__builtin_amdgcn_global_load_async_to_lds_b128)
  // CDNA5 async data-mover path: stream the row straight into LDS
  // (no VGPR load-return), tracked on this wave's ASYNCcnt.
  float* lrow = smem + wave * ROW_LEN;
#pragma unroll
  for (int c = 0; c < CHUNKS; ++c) {
    gv4i_ptr g = (gv4i_ptr)(rowp + c * 128 + lane * 4);
    lv4i_ptr l = (lv4i_ptr)(lrow + c * 128 + lane * 4);
    __builtin_amdgcn_global_load_async_to_lds_b128(g, l, 0, 0);
  }
  // Each wave consumes only the row it loaded itself -> waiting on the
  // per-wave ASYNCcnt is sufficient; no workgroup barrier needed.
#if __has_builtin(__builtin_amdgcn_s_wait_asynccnt)
  __builtin_amdgcn_s_wait_asynccnt(0);
#else
  asm volatile("s_wait_asynccnt 0" ::: "memory");
#endif
#pragma unroll
  for (int c = 0; c < CHUNKS; ++c)
    vals[c] = *(const v4f*)(lrow + c * 128 + lane * 4);
#else
  // Fallback: direct coalesced b128 loads into registers.
#pragma unroll
  for (int c = 0; c < CHUNKS; ++c)
    vals[c] = *(const v4f*)(rowp + c * 128 + lane * 4);
#endif

  // Pass 1: row max (registers + wave32 shuffle reduce).
  float m = -__builtin_inff();
#pragma unroll
  for (int c = 0; c < CHUNKS; ++c) {
    m = fmaxf(m, fmaxf(fmaxf(vals[c].x, vals[c].y), fmaxf(vals[c].z, vals[c].w)));
  }
  m = wave_max(m);

  // Pass 2: exp (hardware v_exp_f32) + sum.
  float s = 0.0f;
#pragma unroll
  for (int c = 0; c < CHUNKS; ++c) {
    v4f e;
    e.x = __expf(vals[c].x - m);
    e.y = __expf(vals[c].y - m);
    e.z = __expf(vals[c].z - m);
    e.w = __expf(vals[c].w - m);
    vals[c] = e;
    s += (e.x + e.y) + (e.z + e.w);
  }
  s = wave_sum(s);

  // Pass 3: scale + non-temporal b128 stores (output is never re-read;
  // keep it out of L2 so the streaming input keeps the cache).
  const float inv = 1.0f / s;
#pragma unroll
  for (int c = 0; c < CHUNKS; ++c) {
    v4f o = vals[c] * inv;
    __builtin_nontemporal_store(o, (v4f*)(orow + c * 128 + lane * 4));
  }
}

extern "C" void kernel_launch(void* const* d_in, const int* in_sizes, int n_in,
                              void* d_out, int out_size, void* d_ws, size_t ws_size,
                              hipStream_t stream) {
  (void)n_in; (void)d_ws; (void)ws_size; (void)out_size;
  const float* x = (const float*)d_in[0];
  float* out = (float*)d_out;
  const int nrows = in_sizes[0] / ROW_LEN;       // 65536
  const int nblocks = (nrows + WAVES_PER_BLK - 1) / WAVES_PER_BLK;  // 8192
  softmax1024_kernel<<<nblocks, THREADS, 0, stream>>>(x, out, nrows);
}